// SiameseGNN_84035330114248
// MI455X (gfx1250) — compile-verified
//
#include <hip/hip_runtime.h>
#include <hip/hip_bf16.h>
#include <math.h>

// ---------------- problem constants (from reference) ----------------
#define N1   128000
#define N2   400
#define Bg   64
#define Ksel 30
#define Hd   128
#define NPG  2000
#define E1   1024000
#define E2   6400

typedef __attribute__((ext_vector_type(2))) float v2f;
typedef __attribute__((ext_vector_type(8))) float v8f;

// ---------------- small helpers ----------------
__global__ void init_ones_kernel(float* p, int n) {
    int i = blockIdx.x * blockDim.x + threadIdx.x;
    if (i < n) p[i] = 1.0f;
}

__global__ void deg_edges_kernel(const int* __restrict__ dst, int E, float* __restrict__ deg) {
    int e = blockIdx.x * blockDim.x + threadIdx.x;
    if (e < E) atomicAdd(&deg[dst[e]], 1.0f);
}

__global__ void make_dinv_kernel(float* __restrict__ deg, int n) {
    int i = blockIdx.x * blockDim.x + threadIdx.x;
    if (i < n) deg[i] = rsqrtf(fmaxf(deg[i], 1e-12f));
}

// layer-1 aggregation collapses to a scalar scatter because x is [N,1]
__global__ void agg1_edges_kernel(const int* __restrict__ src, const int* __restrict__ dst,
                                  const float* __restrict__ x, const float* __restrict__ dinv,
                                  float* __restrict__ s, int E) {
    int e = blockIdx.x * blockDim.x + threadIdx.x;
    if (e < E) {
        int a = src[e], d = dst[e];
        atomicAdd(&s[d], dinv[a] * dinv[d] * x[a]);
    }
}

__global__ void layer1_fin_kernel(const float* __restrict__ s, const float* __restrict__ x,
                                  const float* __restrict__ dinv, const float* __restrict__ W1,
                                  const float* __restrict__ b1, float* __restrict__ h1, int n) {
    long t = (long)blockIdx.x * blockDim.x + threadIdx.x;
    if (t < (long)n * Hd) {
        int i = (int)(t >> 7), c = (int)(t & 127);
        float di = dinv[i];
        float val = s[i] + di * di * x[i];
        h1[t] = fmaxf(val * W1[c] + b1[c], 0.0f);
    }
}

// layer-2 row scatter: agg[dst,:] += norm * t[src,:]
__global__ void agg2_edges_kernel(const int* __restrict__ src, const int* __restrict__ dst,
                                  const float* __restrict__ dinv, const float* __restrict__ t,
                                  float* __restrict__ agg, long EH) {
    long id = (long)blockIdx.x * blockDim.x + threadIdx.x;
    if (id < EH) {
        int e = (int)(id >> 7), c = (int)(id & 127);
        int a = src[e], d = dst[e];
        atomicAdd(&agg[(long)d * Hd + c], dinv[a] * dinv[d] * t[(long)a * Hd + c]);
    }
}

__global__ void layer2_fin_kernel(float* __restrict__ agg, const float* __restrict__ t,
                                  const float* __restrict__ dinv, const float* __restrict__ b2, int n) {
    long id = (long)blockIdx.x * blockDim.x + threadIdx.x;
    if (id < (long)n * Hd) {
        int i = (int)(id >> 7), c = (int)(id & 127);
        float di = dinv[i];
        agg[id] = fmaxf(agg[id] + di * di * t[id] + b2[c], 0.0f);
    }
}

// ---------------- generic f32 WMMA GEMM: C[M,N] = A*B ----------------
// A(i,k)=A[i*lda+k], B(k,j)=B[k*brs+j*bcs]. M,N multiples of 16, K multiple of 4.
// one wave per 16x16 tile; V_WMMA_F32_16X16X4_F32, K-loop step 4.
__global__ void gemm_wmma_f32(const float* __restrict__ A, const float* __restrict__ B,
                              float* __restrict__ C, int K,
                              long lda, long brs, long bcs, long ldc) {
    int tm = blockIdx.x, tn = blockIdx.y;
    int lane = threadIdx.x & 31;
    int half = lane >> 4;
    int mr   = lane & 15;
    long arow = (long)tm * 16 + mr;   // A row this lane feeds
    long bcol = (long)tn * 16 + mr;   // B col this lane feeds
    v8f c = {};
    for (int k0 = 0; k0 < K; k0 += 4) {
        long ka = k0 + 2 * half;
        v2f a, b;
        a.x = A[arow * lda + ka];
        a.y = A[arow * lda + ka + 1];
        b.x = B[ka * brs + bcol * bcs];
        b.y = B[(ka + 1) * brs + bcol * bcs];
        c = __builtin_amdgcn_wmma_f32_16x16x4_f32(false, a, false, b, (short)0, c, false, false);
    }
#pragma unroll
    for (int v = 0; v < 8; ++v) {
        long m = (long)tm * 16 + v + half * 8;
        C[m * ldc + bcol] = c[v];
    }
}

// ---------------- row norms + sort key (distance to out2's last row) ----------------
__global__ void rowsq_kernel(const float* __restrict__ X, int n, float* __restrict__ sq) {
    int wid  = (blockIdx.x * blockDim.x + threadIdx.x) >> 5;
    int lane = threadIdx.x & 31;
    if (wid >= n) return;
    const float* r = X + (long)wid * Hd;
    float s = 0.f;
    for (int c = lane; c < Hd; c += 32) { float v = r[c]; s += v * v; }
    for (int o = 16; o; o >>= 1) s += __shfl_xor(s, o, 32);
    if (lane == 0) sq[wid] = s;
}

__global__ void key_kernel(const float* __restrict__ X, const float* __restrict__ ref,
                           const float* __restrict__ n2last, int n,
                           float* __restrict__ sq, float* __restrict__ key) {
    int wid  = (blockIdx.x * blockDim.x + threadIdx.x) >> 5;
    int lane = threadIdx.x & 31;
    if (wid >= n) return;
    const float* r = X + (long)wid * Hd;
    float s = 0.f, d = 0.f;
    for (int c = lane; c < Hd; c += 32) { float v = r[c]; s += v * v; d += v * ref[c]; }
    for (int o = 16; o; o >>= 1) { s += __shfl_xor(s, o, 32); d += __shfl_xor(d, o, 32); }
    if (lane == 0) {
        sq[wid] = s;
        float d2 = s + *n2last - 2.0f * d;
        key[wid] = sqrtf(fmaxf(d2, 0.0f));
    }
}

// ---------------- per-graph top-K (descending, ties -> smaller index) ----------------
__global__ void topk_kernel(const float* __restrict__ key, int* __restrict__ perm) {
    __shared__ float sk[NPG];
    __shared__ unsigned long long red[256];
    int b = blockIdx.x, tid = threadIdx.x;
    for (int r = tid; r < NPG; r += blockDim.x) sk[r] = key[b * NPG + r];
    __syncthreads();
    for (int k = 0; k < Ksel; ++k) {
        unsigned long long best = 0;
        for (int r = tid; r < NPG; r += blockDim.x) {
            float f = sk[r];
            if (f >= 0.0f) {  // keys are >=0; removed entries marked -1
                unsigned long long p =
                    ((unsigned long long)__float_as_uint(f) << 32) |
                    (unsigned int)(0x7FFFFFFFu - r);
                if (p > best) best = p;
            }
        }
        red[tid] = best;
        __syncthreads();
        for (int s = blockDim.x / 2; s > 0; s >>= 1) {
            if (tid < s && red[tid + s] > red[tid]) red[tid] = red[tid + s];
            __syncthreads();
        }
        if (tid == 0) {
            int r = (int)(0x7FFFFFFFu - (unsigned int)(red[0] & 0xFFFFFFFFu));
            perm[b * Ksel + k] = b * NPG + r;
            sk[r] = -1.0f;
        }
        __syncthreads();
    }
}

// ---------------- gathered-row cdist via WMMA: topd[1920,400] ----------------
__global__ void topk_dist_wmma(const float* __restrict__ out1, const float* __restrict__ out2,
                               const int* __restrict__ perm, const float* __restrict__ n1sq,
                               const float* __restrict__ n2sq, float* __restrict__ topd) {
    int tm = blockIdx.x;            // 0..119 (1920/16)
    int tn = blockIdx.y;            // 0..24  (400/16)
    int lane = threadIdx.x & 31;
    int half = lane >> 4;
    int mr   = lane & 15;
    int gm   = tm * 16 + mr;
    long rowi = perm[gm];
    long col  = (long)tn * 16 + mr;
    v8f c = {};
    for (int k0 = 0; k0 < Hd; k0 += 4) {
        long ka = k0 + 2 * half;
        v2f a, b;
        a.x = out1[rowi * Hd + ka];
        a.y = out1[rowi * Hd + ka + 1];
        b.x = out2[col * Hd + ka];      // B = out2^T  (brs=1, bcs=Hd)
        b.y = out2[col * Hd + ka + 1];
        c = __builtin_amdgcn_wmma_f32_16x16x4_f32(false, a, false, b, (short)0, c, false, false);
    }
    float nn = n2sq[col];
#pragma unroll
    for (int v = 0; v < 8; ++v) {
        int m = tm * 16 + v + half * 8;
        int r = perm[m];
        float d2 = n1sq[r] + nn - 2.0f * c[v];
        topd[(long)m * N2 + col] = sqrtf(fmaxf(d2, 0.0f));
    }
}

// ---------------- bias + layernorm + relu (W = 128 or 64 threads/block) ----------------
__global__ void bias_ln_relu_kernel(const float* __restrict__ X, const float* __restrict__ bias,
                                    const float* __restrict__ g, const float* __restrict__ bb,
                                    float* __restrict__ Y, int W) {
    __shared__ float red[128];
    int row = blockIdx.x, c = threadIdx.x;
    float x = X[row * W + c] + bias[c];
    red[c] = x; __syncthreads();
    for (int s = W >> 1; s > 0; s >>= 1) { if (c < s) red[c] += red[c + s]; __syncthreads(); }
    float mu = red[0] / (float)W;
    __syncthreads();
    float dx = x - mu;
    red[c] = dx * dx; __syncthreads();
    for (int s = W >> 1; s > 0; s >>= 1) { if (c < s) red[c] += red[c + s]; __syncthreads(); }
    float var = red[0] / (float)W;
    float y = dx * rsqrtf(var + 1e-5f) * g[c] + bb[c];
    Y[row * W + c] = fmaxf(y, 0.0f);
}

__global__ void fc3_sigmoid_kernel(const float* __restrict__ H2, const float* __restrict__ W,
                                   const float* __restrict__ b, float* __restrict__ out) {
    __shared__ float red[64];
    int row = blockIdx.x, c = threadIdx.x;
    red[c] = H2[row * 64 + c] * W[c]; __syncthreads();
    for (int s = 32; s > 0; s >>= 1) { if (c < s) red[c] += red[c + s]; __syncthreads(); }
    if (c == 0) {
        float z = red[0] + b[0];
        out[row] = 1.0f / (1.0f + expf(-z));
    }
}

// ---------------- host orchestration ----------------
static void run_gnn(const float* x, const int* ei, int n, int E,
                    const float* W1, const float* b1, const float* W2, const float* b2,
                    float* dinv, float* s, float* bufA /*h1 -> agg -> out*/, float* bufT,
                    hipStream_t stream) {
    const int* src = ei;
    const int* dst = ei + E;
    init_ones_kernel<<<(n + 255) / 256, 256, 0, stream>>>(dinv, n);
    deg_edges_kernel<<<(E + 255) / 256, 256, 0, stream>>>(dst, E, dinv);
    make_dinv_kernel<<<(n + 255) / 256, 256, 0, stream>>>(dinv, n);
    hipMemsetAsync(s, 0, (size_t)n * sizeof(float), stream);
    agg1_edges_kernel<<<(E + 255) / 256, 256, 0, stream>>>(src, dst, x, dinv, s, E);
    long nH = (long)n * Hd;
    layer1_fin_kernel<<<(int)((nH + 255) / 256), 256, 0, stream>>>(s, x, dinv, W1, b1, bufA, n);
    // t = h1 @ W2   (f32 WMMA)
    gemm_wmma_f32<<<dim3(n / 16, Hd / 16), 32, 0, stream>>>(bufA, W2, bufT, Hd, Hd, Hd, 1, Hd);
    hipMemsetAsync(bufA, 0, (size_t)nH * sizeof(float), stream);   // reuse h1 buffer as accumulator
    long EH = (long)E * Hd;
    agg2_edges_kernel<<<(int)((EH + 255) / 256), 256, 0, stream>>>(src, dst, dinv, bufT, bufA, EH);
    layer2_fin_kernel<<<(int)((nH + 255) / 256), 256, 0, stream>>>(bufA, bufT, dinv, b2, n);
}

extern "C" void kernel_launch(void* const* d_in, const int* in_sizes, int n_in,
                              void* d_out, int out_size, void* d_ws, size_t ws_size,
                              hipStream_t stream) {
    const float* x1   = (const float*)d_in[0];
    const float* x2   = (const float*)d_in[1];
    const int*   ei1  = (const int*)d_in[2];
    const int*   ei2  = (const int*)d_in[3];
    // d_in[4] = batch1 (unused: equal-size graphs, arange//NPG)
    const float* W1   = (const float*)d_in[5];
    const float* b1   = (const float*)d_in[6];
    const float* W2   = (const float*)d_in[7];
    const float* b2   = (const float*)d_in[8];
    const float* fc1W = (const float*)d_in[9];
    const float* fc1b = (const float*)d_in[10];
    const float* ln1g = (const float*)d_in[11];
    const float* ln1b = (const float*)d_in[12];
    const float* fc2W = (const float*)d_in[13];
    const float* fc2b = (const float*)d_in[14];
    const float* ln2g = (const float*)d_in[15];
    const float* ln2b = (const float*)d_in[16];
    const float* fc3W = (const float*)d_in[17];
    const float* fc3b = (const float*)d_in[18];

    // bump allocator on workspace
    char* wsp = (char*)d_ws;
    auto alloc = [&](size_t bytes) -> void* {
        void* p = (void*)wsp;
        wsp += (bytes + 255) & ~(size_t)255;
        return p;
    };
    float* dinv1 = (float*)alloc((size_t)N1 * 4);
    float* s1    = (float*)alloc((size_t)N1 * 4);
    float* bufA1 = (float*)alloc((size_t)N1 * Hd * 4);   // h1 / agg / out1
    float* bufT1 = (float*)alloc((size_t)N1 * Hd * 4);   // t1
    float* dinv2 = (float*)alloc((size_t)N2 * 4);
    float* s2    = (float*)alloc((size_t)N2 * 4);
    float* bufA2 = (float*)alloc((size_t)N2 * Hd * 4);   // out2 lives here
    float* bufT2 = (float*)alloc((size_t)N2 * Hd * 4);
    float* n1sq  = (float*)alloc((size_t)N1 * 4);
    float* keyb  = (float*)alloc((size_t)N1 * 4);
    float* n2sq  = (float*)alloc((size_t)N2 * 4);
    int*   perm  = (int*)  alloc((size_t)Bg * Ksel * 4);
    float* topd  = (float*)alloc((size_t)Bg * Ksel * N2 * 4);  // [64, 12000]
    float* m1    = (float*)alloc((size_t)Bg * 128 * 4);
    float* m1a   = (float*)alloc((size_t)Bg * 128 * 4);
    float* m2    = (float*)alloc((size_t)Bg * 64 * 4);
    float* m2a   = (float*)alloc((size_t)Bg * 64 * 4);

    // two-branch siamese GNN
    run_gnn(x1, ei1, N1, E1, W1, b1, W2, b2, dinv1, s1, bufA1, bufT1, stream);
    run_gnn(x2, ei2, N2, E2, W1, b1, W2, b2, dinv2, s2, bufA2, bufT2, stream);

    // row norms of out2, then sort key = d(out1_i, out2_last) for every node in graph1
    rowsq_kernel<<<(N2 + 7) / 8, 256, 0, stream>>>(bufA2, N2, n2sq);
    key_kernel<<<(N1 + 7) / 8, 256, 0, stream>>>(bufA1, bufA2 + (size_t)(N2 - 1) * Hd,
                                                 n2sq + (N2 - 1), N1, n1sq, keyb);

    // per-graph top-30 row selection (stable-argsort tie-break)
    topk_kernel<<<Bg, 256, 0, stream>>>(keyb, perm);

    // distances for only the selected 1920 rows (WMMA, out2 transposed access)
    topk_dist_wmma<<<dim3((Bg * Ksel) / 16, N2 / 16), 32, 0, stream>>>(
        bufA1, bufA2, perm, n1sq, n2sq, topd);

    // MLP head: fc1 (WMMA, K=12000) -> LN -> relu -> fc2 (WMMA) -> LN -> relu -> fc3 -> sigmoid
    gemm_wmma_f32<<<dim3(Bg / 16, 128 / 16), 32, 0, stream>>>(topd, fc1W, m1,
                                                              Ksel * N2, Ksel * N2, 128, 1, 128);
    bias_ln_relu_kernel<<<Bg, 128, 0, stream>>>(m1, fc1b, ln1g, ln1b, m1a, 128);
    gemm_wmma_f32<<<dim3(Bg / 16, 64 / 16), 32, 0, stream>>>(m1a, fc2W, m2,
                                                             128, 128, 64, 1, 64);
    bias_ln_relu_kernel<<<Bg, 64, 0, stream>>>(m2, fc2b, ln2g, ln2b, m2a, 64);
    fc3_sigmoid_kernel<<<Bg, 64, 0, stream>>>(m2a, fc3W, fc3b, (float*)d_out);
    (void)in_sizes; (void)n_in; (void)out_size; (void)ws_size;
}